// TopKPooling_6983616823980
// MI455X (gfx1250) — compile-verified
//
#include <hip/hip_runtime.h>
#include <math.h>

#define B_   32
#define N_   8192
#define F_   256
#define K_   4096

typedef __attribute__((ext_vector_type(2))) float v2f;
typedef __attribute__((ext_vector_type(8))) float v8f;

// ---------------------------------------------------------------------------
// Kernel 1: score[b,n] = tanh( (x[b,n,:] . w) / ||w|| )  via V_WMMA_F32_16X16X4_F32
// One wave computes 16 rows; 8 waves/block -> 128 rows/block.
// A (16x4 f32): lane<16 -> M=lane, K = k+{0,1}; lane>=16 -> M=lane-16, K = k+{2,3}
// B (4x16 f32): all 16 columns replicate w, same per-lane K split as A.
// D (16x16 f32): lane0 VGPR v = row v; lane16 VGPR v = row v+8 (column 0).
// ---------------------------------------------------------------------------
__global__ __launch_bounds__(256) void topk_score_kernel(
    const float* __restrict__ x, const float* __restrict__ w,
    float* __restrict__ scores)
{
    __shared__ float wlds[F_];
    __shared__ float s_invnorm;

    const int tid = threadIdx.x;
    wlds[tid] = w[tid];               // blockDim.x == F_ == 256
    __syncthreads();
    if (tid == 0) {
        float s = 0.f;
        #pragma unroll 8
        for (int i = 0; i < F_; ++i) s += wlds[i] * wlds[i];
        s_invnorm = rsqrtf(s);
    }
    __syncthreads();

    const int wave = tid >> 5;
    const int lane = tid & 31;
    const long rowBase = ((long)blockIdx.x * 8 + wave) * 16;
    const int  m    = lane & 15;           // row within 16-row tile
    const int  koff = (lane >> 4) * 2;     // 0 for lanes 0-15, 2 for lanes 16-31

    const float* xrow = x + (rowBase + m) * F_ + koff;
    const float* wrow = wlds + koff;

    v8f acc = {};
    #pragma unroll 8
    for (int k = 0; k < F_; k += 4) {
        v2f a = *(const v2f*)(xrow + k);   // x[row, k+koff .. k+koff+1]
        v2f b = *(const v2f*)(wrow + k);   // w[k+koff .. k+koff+1] (all columns equal)
        acc = __builtin_amdgcn_wmma_f32_16x16x4_f32(
            /*neg_a=*/false, a, /*neg_b=*/false, b,
            /*c_mod=*/(short)0, acc, /*reuse_a=*/false, /*reuse_b=*/false);
    }

    const float inv = s_invnorm;
    if (lane == 0 || lane == 16) {
        const long o = rowBase + ((lane == 16) ? 8 : 0);
        #pragma unroll
        for (int i = 0; i < 8; ++i)
            scores[o + i] = tanhf(acc[i] * inv);
    }
}

// ---------------------------------------------------------------------------
// Kernel 2: per-batch bitonic sort of 8192 (key,idx) packed u64 in LDS.
// key = order-flipped float bits, inverted so ascending u64 == descending score
// (ties -> smaller index first, matching stable argsort(-score)).
// Emits perm[b,k] and the selected score (recovered from the key bits).
// ---------------------------------------------------------------------------
__global__ __launch_bounds__(1024) void topk_sort_kernel(
    const float* __restrict__ scores,
    int* __restrict__ perm, float* __restrict__ selscore)
{
    __shared__ unsigned long long s[N_];   // 64 KB
    const int b = blockIdx.x;
    const float* sc = scores + (long)b * N_;

    for (int i = threadIdx.x; i < N_; i += 1024) {
        unsigned u = __float_as_uint(sc[i]);
        u = (u & 0x80000000u) ? ~u : (u | 0x80000000u); // monotone float->uint
        u = ~u;                                         // descending -> ascending
        s[i] = ((unsigned long long)u << 32) | (unsigned)i;
    }

    for (int kk = 2; kk <= N_; kk <<= 1) {
        for (int j = kk >> 1; j > 0; j >>= 1) {
            __syncthreads();
            for (int t = threadIdx.x; t < (N_ / 2); t += 1024) {
                const int i = ((t & ~(j - 1)) << 1) | (t & (j - 1));
                const int p = i | j;
                const unsigned long long a = s[i];
                const unsigned long long c = s[p];
                const bool up = ((i & kk) == 0);
                if ((a > c) == up) { s[i] = c; s[p] = a; }
            }
        }
    }
    __syncthreads();

    for (int i = threadIdx.x; i < K_; i += 1024) {
        const unsigned long long v = s[i];
        const unsigned idx = (unsigned)v;
        unsigned u = ~(unsigned)(v >> 32);              // un-invert
        u = (u & 0x80000000u) ? (u ^ 0x80000000u) : ~u; // un-flip
        perm[(long)b * K_ + i] = (int)idx;
        selscore[(long)b * K_ + i] = __uint_as_float(u);
    }
}

// ---------------------------------------------------------------------------
// Kernel 3: gather + scale.  4 rows per 256-thread block, float4 vectorized.
// x_out[b,k,:] = x[b,perm[b,k],:] * selscore[b,k];  pos_out[b,k,:] = posi[...]
// ---------------------------------------------------------------------------
__global__ __launch_bounds__(256) void topk_gather_kernel(
    const float* __restrict__ x, const float* __restrict__ posi,
    const int* __restrict__ perm, const float* __restrict__ selscore,
    float* __restrict__ xout, float* __restrict__ posout)
{
    const long row = (long)blockIdx.x * 4 + (threadIdx.x >> 6);  // [0, B*K)
    const int  f4  = threadIdx.x & 63;                           // 64 float4 per row
    const long b   = row >> 12;                                  // row / K_
    const int  idx = perm[row];
    const float sc = selscore[row];

    const long srcRow = b * N_ + idx;
    const float4 v = ((const float4*)(x + srcRow * F_))[f4];
    float4 o;
    o.x = v.x * sc; o.y = v.y * sc; o.z = v.z * sc; o.w = v.w * sc;
    ((float4*)(xout + row * F_))[f4] = o;

    if (f4 < 3)
        posout[row * 3 + f4] = posi[srcRow * 3 + f4];
}

// ---------------------------------------------------------------------------
extern "C" void kernel_launch(void* const* d_in, const int* in_sizes, int n_in,
                              void* d_out, int out_size, void* d_ws, size_t ws_size,
                              hipStream_t stream) {
    const float* x    = (const float*)d_in[0];   // [B,N,F]
    const float* posi = (const float*)d_in[1];   // [B,N,3]
    const float* w    = (const float*)d_in[2];   // [1,F]

    float* xout   = (float*)d_out;                        // [B,K,F]
    float* posout = xout + (size_t)B_ * K_ * F_;          // [B,K,3]

    float* scores   = (float*)d_ws;                       // B*N floats
    int*   perm     = (int*)(scores + (size_t)B_ * N_);   // B*K ints
    float* selscore = (float*)(perm + (size_t)B_ * K_);   // B*K floats

    topk_score_kernel<<<(B_ * N_) / 128, 256, 0, stream>>>(x, w, scores);
    topk_sort_kernel<<<B_, 1024, 0, stream>>>(scores, perm, selscore);
    topk_gather_kernel<<<(B_ * K_) / 4, 256, 0, stream>>>(x, posi, perm, selscore,
                                                          xout, posout);
}